// GSAB_5420248727614
// MI455X (gfx1250) — compile-verified
//
#include <hip/hip_runtime.h>

// CAVE sigmoid-family backward: per-row reduction over x[B,N], N=16384, B=4096.
// Memory-bound: 256 MiB of x @ 23.3 TB/s => ~11.5us floor. Element math:
// sigmoid(z)=0.5+0.5*tanh(z/2) -> 1 trans + ~10 VALU (compiler packs to
// v_pk_/v_dual_ f32). Wave-level 6-way reduction via V_WMMA_F32_16X16X4_F32.

typedef float v2f __attribute__((ext_vector_type(2)));
typedef float v8f __attribute__((ext_vector_type(8)));

#if __has_builtin(__builtin_amdgcn_tanhf)
#define HAVE_HW_TANH 1
#else
#warning "CDNA5 probe: __builtin_amdgcn_tanhf NOT available; using exp2+rcp sigmoid"
#endif

#if __has_builtin(__builtin_amdgcn_wmma_f32_16x16x4_f32)
#define HAVE_WMMA_F32 1
#else
#warning "CDNA5 probe: wmma_f32_16x16x4_f32 NOT available; using shfl reduction"
#endif

// Reduce 6 per-lane partials to full-wave(32) sums, broadcast to all lanes.
__device__ __forceinline__ void wave_reduce6(float v[6]) {
#if defined(HAVE_WMMA_F32)
  v2f ones; ones[0] = 1.0f; ones[1] = 1.0f;
  v8f zero = {};
  v8f d[6];
#pragma unroll
  for (int q = 0; q < 6; ++q) {
    v2f am; am[0] = v[q]; am[1] = 0.0f;
    // D1[m,n] = s[m] + s[m+16]  (A = data, B = ones)
    d[q] = __builtin_amdgcn_wmma_f32_16x16x4_f32(
        false, am, false, ones, (short)0, zero, false, false);
  }
#pragma unroll
  for (int q = 0; q < 6; ++q) {
    v2f bm;
    bm[0] = (d[q][0] + d[q][2]) + (d[q][4] + d[q][6]);
    bm[1] = (d[q][1] + d[q][3]) + (d[q][5] + d[q][7]);
    // D2[m,n] = sum over all 4 K slots x both lane halves = full 32-lane sum
    v8f d2 = __builtin_amdgcn_wmma_f32_16x16x4_f32(
        false, ones, false, bm, (short)0, zero, false, false);
    v[q] = d2[0];
  }
#else
#pragma unroll
  for (int q = 0; q < 6; ++q) {
    float s = v[q];
#pragma unroll
    for (int off = 16; off > 0; off >>= 1) s += __shfl_xor(s, off, 32);
    v[q] = s;
  }
#endif
}

__device__ __forceinline__ void accum_elem(float xe, float ah, float bh,
                                           float& s1, float& s2, float& s3,
                                           float& s4, float& s5, float& s6) {
  float t = __builtin_fmaf(ah, xe, bh);          // z/2
#if defined(HAVE_HW_TANH)
  float th = __builtin_amdgcn_tanhf(t);          // v_tanh_f32 (1 trans op)
  float s  = __builtin_fmaf(0.5f, th, 0.5f);     // sigmoid(z)
  float sp = __builtin_fmaf(-0.25f * th, th, 0.25f); // s*(1-s) = (1-th^2)/4
#else
  float e  = __builtin_amdgcn_exp2f(t * -2.885390081777927f); // exp(-z)
  float s  = __builtin_amdgcn_rcpf(1.0f + e);
  float sp = (1.0f - s) * s;
#endif
  float xsp = xe * sp;
  s1 += s;
  s2 += xsp;
  s3 += sp;
  s4 = __builtin_fmaf(s, s, s4);
  s5 = __builtin_fmaf(s, xsp, s5);
  s6 = __builtin_fmaf(s, sp, s6);
}

template <int ITERS>
__global__ void __launch_bounds__(256)
cave_sigmoid_bwd(const float* __restrict__ x, const float* __restrict__ a,
                 const float* __restrict__ b, const float* __restrict__ mean,
                 const float* __restrict__ var, float* __restrict__ out,
                 int B, int N, int iters_dyn) {
  const int row = blockIdx.x;
  const int tid = threadIdx.x;

  const float ah = 0.5f * a[row];
  const float bh = 0.5f * b[row];
  const float4* __restrict__ xr =
      (const float4*)(x + (size_t)row * (size_t)N);

  float s1 = 0.f, s2 = 0.f, s3 = 0.f, s4 = 0.f, s5 = 0.f, s6 = 0.f;

  const int iters = (ITERS > 0) ? ITERS : iters_dyn;
#pragma unroll
  for (int i = 0; i < iters; ++i) {
    float4 xv = xr[i * 256 + tid];  // coalesced global_load_b128
    accum_elem(xv.x, ah, bh, s1, s2, s3, s4, s5, s6);
    accum_elem(xv.y, ah, bh, s1, s2, s3, s4, s5, s6);
    accum_elem(xv.z, ah, bh, s1, s2, s3, s4, s5, s6);
    accum_elem(xv.w, ah, bh, s1, s2, s3, s4, s5, s6);
  }

  float sums[6] = {s1, s2, s3, s4, s5, s6};
  wave_reduce6(sums);  // every lane now holds its wave's totals

  __shared__ float red[8][6];
  const int lane = tid & 31;
  const int wave = tid >> 5;
  if (lane == 0) {
#pragma unroll
    for (int q = 0; q < 6; ++q) red[wave][q] = sums[q];
  }
  __syncthreads();

  if (tid == 0) {
    float T[6];
#pragma unroll
    for (int q = 0; q < 6; ++q) {
      float acc = red[0][q];
#pragma unroll
      for (int w = 1; w < 8; ++w) acc += red[w][q];
      T[q] = acc;
    }
    // N is a power of two (16384) => v_rcp is exact.
    const float Minv = __builtin_amdgcn_rcpf((float)N);
    float fm     = T[0] * Minv;
    float em     = fm - mean[row];
    float dem_da = T[1] * Minv;
    float dem_db = T[2] * Minv;
    float ev     = __builtin_fmaf(-fm, fm, T[3] * Minv) - var[row];
    float dev_da = 2.0f * __builtin_fmaf(-fm, dem_da, T[4] * Minv);
    float dev_db = 2.0f * __builtin_fmaf(-fm, dem_db, T[5] * Minv);
    out[row]     = 2.0f * (em * dem_da + ev * dev_da);  // dl_da
    out[B + row] = 2.0f * (em * dem_db + ev * dev_db);  // dl_db
  }
}

extern "C" void kernel_launch(void* const* d_in, const int* in_sizes, int n_in,
                              void* d_out, int out_size, void* d_ws,
                              size_t ws_size, hipStream_t stream) {
  (void)n_in; (void)out_size; (void)d_ws; (void)ws_size;
  const float* x    = (const float*)d_in[0];
  const float* a    = (const float*)d_in[1];
  const float* b    = (const float*)d_in[2];
  const float* mean = (const float*)d_in[3];
  const float* var  = (const float*)d_in[4];
  float* out = (float*)d_out;

  const int B = in_sizes[1];           // a is [B,1]
  const int N = in_sizes[0] / B;       // x is [B,N]

  if (N == 16384) {
    // Compile-time trip count: full unroll, 16 b128 loads in flight, no
    // remainder control flow.
    cave_sigmoid_bwd<16><<<B, 256, 0, stream>>>(x, a, b, mean, var, out, B, N, 0);
  } else {
    cave_sigmoid_bwd<0><<<B, 256, 0, stream>>>(x, a, b, mean, var, out, B, N,
                                               N >> 10);
  }
}